// SSD_64424509440030
// MI455X (gfx1250) — compile-verified
//
#include <hip/hip_runtime.h>
#include <hip/hip_bf16.h>
#include <stdint.h>

#define NUM_PRIORS 24532
#define BATCH      64
#define NCLS       21
#define TOPK       200
#define CONF_TH    0.01f
#define NMS_TH     0.45f
#define CHUNK      4096

typedef float v2f  __attribute__((ext_vector_type(2)));
typedef float v8f  __attribute__((ext_vector_type(8)));
typedef unsigned int u32x4 __attribute__((ext_vector_type(4)));
typedef int  i32x4 __attribute__((ext_vector_type(4)));
typedef int  i32x8 __attribute__((ext_vector_type(8)));

#if __has_builtin(__builtin_amdgcn_tensor_load_to_lds)
#define HAVE_TDM 1
#else
#define HAVE_TDM 0
#endif
#if __has_builtin(__builtin_amdgcn_wmma_f32_16x16x4_f32)
#define HAVE_WMMA4 1
#else
#define HAVE_WMMA4 0
#endif

#if HAVE_TDM
// DMA a strided column (stride_elems f32 apart) of `rows` elements into LDS,
// packed contiguously. D# built per CDNA5 ISA ch.8: 2-D tile, tile_dim0=1,
// tile_dim1=rows, tensor_dim0_stride=stride. Groups 2/3/4 zero (dims unused).
// This toolchain (clang-23 / therock-10.0 headers) uses the 6-arg form:
//   (uint32x4 g0, int32x8 g1, int32x4 g2, int32x4 g3, int32x8 g4, i32 cpol)
__device__ __forceinline__ void tdm_load_strided_col(const float* gsrc,
                                                     unsigned lds_byte_off,
                                                     int rows, int stride_elems) {
  unsigned long long ga = (unsigned long long)(uintptr_t)gsrc;
  u32x4 g0 = {
      1u,                                   // count=1, user mode, no gather
      lds_byte_off,                         // lds_addr
      (unsigned)ga,                         // global_addr[31:0]
      ((unsigned)(ga >> 32) & 0x01FFFFFFu) | (2u << 30)  // addr[56:32] | type=2
  };
  const unsigned td0 = 64u;                 // tensor_dim0 (>= tile_dim0, no clip)
  const unsigned td1 = 0x00FFFFFFu;         // tensor_dim1 large (bounds in SW)
  i32x8 g1 = {
      (int)(2u << 16),                                          // data_size=4B
      (int)((td0 & 0xFFFFu) << 16),                             // dim0[15:0]
      (int)(((td0 >> 16) & 0xFFFFu) | ((td1 & 0xFFFFu) << 16)), // dim0 hi|dim1 lo
      (int)(((td1 >> 16) & 0xFFFFu) | (1u << 16)),              // dim1 hi|tile_dim0=1
      (int)((unsigned)rows & 0xFFFFu),                          // tile_dim1=rows
      (int)(unsigned)stride_elems,                              // dim0_stride lo
      0, 0
  };
  i32x4 zz4 = {0, 0, 0, 0};
  i32x8 zz8 = {0, 0, 0, 0, 0, 0, 0, 0};
  __builtin_amdgcn_tensor_load_to_lds(g0, g1, zz4, zz4, zz8, 0);
}
#endif

__device__ __forceinline__ void load_col_chunk(const float* col_base, float* buf,
                                               int rows) {
#if HAVE_TDM
  if ((threadIdx.x >> 5) == 0) {            // one DMA per block (wave 0)
    tdm_load_strided_col(col_base, (unsigned)(uintptr_t)buf, rows, NCLS);
    __builtin_amdgcn_s_wait_tensorcnt(0);
  }
  __syncthreads();
#else
  for (int i = threadIdx.x; i < rows; i += blockDim.x)
    buf[i] = col_base[(size_t)i * NCLS];
  __syncthreads();
#endif
}

// ---------------- Kernel 1: exact top-200 per (batch, class) -----------------
__global__ __launch_bounds__(256) void topk_kernel(const float* __restrict__ conf,
                                                   float* __restrict__ tsc,
                                                   int* __restrict__ tix) {
  const int ch  = blockIdx.x + 1;   // conf channel 1..20 (skip background)
  const int b   = blockIdx.y;
  const int tid = threadIdx.x;

  __shared__ float    buf[CHUNK];
  __shared__ unsigned hist[256];
  __shared__ unsigned skey[256];
  __shared__ int      sidx[256];
  __shared__ unsigned sh_prefix;
  __shared__ int      sh_rem;
  __shared__ unsigned s_gt, s_eq;

  const float* colb = conf + (size_t)b * NUM_PRIORS * NCLS + ch;

  if (tid == 0) { sh_prefix = 0u; sh_rem = TOPK; }
  __syncthreads();

  // 4-pass MSB radix selection on order-preserving uint keys.
  for (int pass = 3; pass >= 0; --pass) {
    const int shift = pass * 8;
    const unsigned himask = (pass == 3) ? 0u : (0xFFFFFFFFu << (shift + 8));
    for (int i = tid; i < 256; i += 256) hist[i] = 0u;
    __syncthreads();
    const unsigned pref = sh_prefix;
    for (int c0 = 0; c0 < NUM_PRIORS; c0 += CHUNK) {
      const int rows = min(CHUNK, NUM_PRIORS - c0);
      load_col_chunk(colb + (size_t)c0 * NCLS, buf, rows);
      for (int i = tid; i < rows; i += 256) {
        const float s = buf[i];
        const unsigned key = (s > CONF_TH) ? (__float_as_uint(s) ^ 0x80000000u) : 0u;
        if ((key & himask) == (pref & himask))
          atomicAdd(&hist[(key >> shift) & 255u], 1u);
      }
      __syncthreads();
    }
    if (tid == 0) {
      const int r = sh_rem;
      unsigned c = 0u;
      int bin = 255;
      for (; bin > 0; --bin) {
        const unsigned h = hist[bin];
        if (c + h >= (unsigned)r) break;
        c += h;
      }
      sh_rem    = r - (int)c;
      sh_prefix = pref | ((unsigned)bin << shift);
    }
    __syncthreads();
  }

  const unsigned T   = sh_prefix;   // 200th-largest key
  const int      rem = sh_rem;      // how many ==T to take
  const int      G   = TOPK - rem;  // count of keys strictly > T

  if (tid == 0) { s_gt = 0u; s_eq = 0u; }
  __syncthreads();

  // Compact the 200 winners into skey/sidx.
  for (int c0 = 0; c0 < NUM_PRIORS; c0 += CHUNK) {
    const int rows = min(CHUNK, NUM_PRIORS - c0);
    load_col_chunk(colb + (size_t)c0 * NCLS, buf, rows);
    for (int i = tid; i < rows; i += 256) {
      const float s = buf[i];
      const unsigned key = (s > CONF_TH) ? (__float_as_uint(s) ^ 0x80000000u) : 0u;
      if (key > T) {
        const unsigned pos = atomicAdd(&s_gt, 1u);
        if (pos < (unsigned)G) { skey[pos] = key; sidx[pos] = c0 + i; }
      } else if (key == T) {
        const unsigned q = atomicAdd(&s_eq, 1u);
        if (q < (unsigned)rem) { skey[G + q] = key; sidx[G + q] = c0 + i; }
      }
    }
    __syncthreads();
  }

  if (tid >= TOPK) { skey[tid] = 0u; sidx[tid] = 0x7FFFFFFF; }
  __syncthreads();

  // Bitonic sort 256 elems: key descending, index ascending on ties
  // (reproduces jax.lax.top_k ordering).
  for (unsigned k = 2; k <= 256; k <<= 1) {
    for (unsigned j = k >> 1; j > 0; j >>= 1) {
      const unsigned p = tid ^ j;
      if (p > (unsigned)tid) {
        const unsigned ka = skey[tid], kb = skey[p];
        const int ia = sidx[tid], ib = sidx[p];
        const bool aFirst  = (ka > kb) || (ka == kb && ia < ib);
        const bool descBlk = ((tid & k) == 0);
        if (descBlk ? !aFirst : aFirst) {
          skey[tid] = kb; sidx[tid] = ib;
          skey[p]   = ka; sidx[p]   = ia;
        }
      }
      __syncthreads();
    }
  }

  if (tid < TOPK) {
    const unsigned key = skey[tid];
    const float s = (key == 0u) ? -__builtin_inff()
                                : __uint_as_float(key ^ 0x80000000u);
    const size_t o = (size_t)(b * (NCLS - 1) + (ch - 1)) * TOPK + tid;
    tsc[o] = s;
    tix[o] = sidx[tid];
  }
}

// -------- Kernel 2: gather+decode, IoU (WMMA union), greedy NMS, pack --------
__global__ __launch_bounds__(256) void nms_kernel(const float* __restrict__ loc,
                                                  const float* __restrict__ priors,
                                                  const float* __restrict__ tsc,
                                                  const int* __restrict__ tix,
                                                  float* __restrict__ out) {
  const int cls = blockIdx.x;     // 0..20 (0 = background row, stays zero)
  const int b   = blockIdx.y;
  const int tid = threadIdx.x;

  float* obase = out + (size_t)(b * NCLS + cls) * TOPK * 5;
  if (cls == 0) {
    for (int i = tid; i < TOPK * 5; i += 256) obase[i] = 0.0f;
    return;
  }

  __shared__ float sc[208], bx1[208], by1[208], bx2[208], by2[208], bar[208];
  __shared__ unsigned over[208 * 7];
  __shared__ int slot[TOPK];
  __shared__ int s_cnt;

  const size_t tb = (size_t)(b * (NCLS - 1) + (cls - 1)) * TOPK;

  for (int t = tid; t < 208; t += 256) {
    if (t < TOPK) {
      const float s = tsc[tb + t];
      const int   i = tix[tb + t];
      sc[t] = s;
      const float4 l  = ((const float4*)loc)[(size_t)b * NUM_PRIORS + i];
      const float4 pr = ((const float4*)priors)[i];
      const float cx = pr.x + l.x * 0.1f * pr.z;
      const float cy = pr.y + l.y * 0.1f * pr.w;
      const float w  = pr.z * expf(l.z * 0.2f);
      const float h  = pr.w * expf(l.w * 0.2f);
      const float x1 = cx - 0.5f * w, y1 = cy - 0.5f * h;
      const float x2 = x1 + w,        y2 = y1 + h;
      bx1[t] = x1; by1[t] = y1; bx2[t] = x2; by2[t] = y2;
      bar[t] = (x2 - x1) * (y2 - y1);
    } else {                        // pad rows 200..207: never overlap anything
      sc[t] = -__builtin_inff();
      bx1[t] = 4e9f; by1[t] = 4e9f; bx2[t] = 4e9f; by2[t] = 4e9f;
      bar[t] = 0.0f;
    }
  }
  for (int t = tid; t < 208 * 7; t += 256) over[t] = 0u;
  __syncthreads();

  // 13x13 tiles of 16x16; union term area_m + area_n as exact rank-2 WMMA:
  // A[m][*] = [area_m, 1, 0, 0],  B[*][n] = [1; area_n; 0; 0].
  const int  wv   = tid >> 5;
  const int  lane = tid & 31;
  const bool lo   = lane < 16;
  for (int tt = wv; tt < 13 * 13; tt += 8) {
    const int ti = tt / 13, tj = tt % 13;
    const int n  = tj * 16 + (lane & 15);
    const float an = bar[n];
#if HAVE_WMMA4
    v2f A, Bv;
    A.x  = lo ? bar[ti * 16 + lane] : 0.0f;  // A 16x4: lanes0-15 K=0..1, hi K=2..3
    A.y  = lo ? 1.0f : 0.0f;
    Bv.x = lo ? 1.0f : 0.0f;                 // B 4x16: VGPR0 K0(lo)/K2(hi), VGPR1 K1/K3
    Bv.y = lo ? an   : 0.0f;
    v8f C = {0.f, 0.f, 0.f, 0.f, 0.f, 0.f, 0.f, 0.f};
    v8f D = __builtin_amdgcn_wmma_f32_16x16x4_f32(false, A, false, Bv,
                                                  (short)0, C, false, false);
#endif
    const float xn1 = bx1[n], yn1 = by1[n], xn2 = bx2[n], yn2 = by2[n];
#pragma unroll
    for (int g = 0; g < 8; ++g) {
      const int m = ti * 16 + g + (lo ? 0 : 8);   // D vgpr g: M=g (lo) / 8+g (hi)
#if HAVE_WMMA4
      const float asum = D[g];
#else
      const float asum = bar[m] + an;
#endif
      float xx1 = fmaxf(bx1[m], xn1);
      float yy1 = fmaxf(by1[m], yn1);
      float xx2 = fminf(bx2[m], xn2);
      float yy2 = fminf(by2[m], yn2);
      float iw = xx2 - xx1; iw = iw > 0.f ? iw : 0.f;
      float ih = yy2 - yy1; ih = ih > 0.f ? ih : 0.f;
      const float inter = iw * ih;
      const float uni   = asum - inter;           // union = a_m + a_n - inter
      // inter/uni > t  <=>  inter > t*uni  (uni>0 for any overlappable pair;
      // degenerate uni==0 => inter==0 => false, matching NaN>t == false).
      if (inter > NMS_TH * uni)
        atomicOr(&over[m * 7 + (n >> 5)], 1u << (n & 31));
    }
  }
  __syncthreads();

  // Greedy NMS (matrix is exactly symmetric -> row test == reference column test).
  if (tid == 0) {
    unsigned kw[7] = {0, 0, 0, 0, 0, 0, 0};
    int cnt = 0;
    for (int i = 0; i < TOPK; ++i) {
      unsigned sup = 0u;
      const unsigned* row = &over[i * 7];
#pragma unroll
      for (int t = 0; t < 7; ++t) sup |= kw[t] & row[t];
      const bool valid = sc[i] > CONF_TH;
      if (valid && sup == 0u) { kw[i >> 5] |= 1u << (i & 31); slot[i] = cnt++; }
      else                    { slot[i] = -1; }
    }
    s_cnt = cnt;
  }
  __syncthreads();

  for (int t = tid; t < TOPK; t += 256) {
    const int s = slot[t];
    if (s >= 0) {
      float* o = obase + (size_t)s * 5;
      o[0] = sc[t]; o[1] = bx1[t]; o[2] = by1[t]; o[3] = bx2[t]; o[4] = by2[t];
    }
  }
  for (int k5 = s_cnt * 5 + tid; k5 < TOPK * 5; k5 += 256) obase[k5] = 0.0f;
}

extern "C" void kernel_launch(void* const* d_in, const int* in_sizes, int n_in,
                              void* d_out, int out_size, void* d_ws, size_t ws_size,
                              hipStream_t stream) {
  (void)in_sizes; (void)n_in; (void)out_size; (void)ws_size;
  const float* loc    = (const float*)d_in[0];
  const float* conf   = (const float*)d_in[1];
  const float* priors = (const float*)d_in[2];
  float* out = (float*)d_out;

  float* tsc = (float*)d_ws;                                     // [64*20*200] f32
  int*   tix = (int*)((char*)d_ws +
                      sizeof(float) * BATCH * (NCLS - 1) * TOPK); // [64*20*200] i32

  topk_kernel<<<dim3(NCLS - 1, BATCH), 256, 0, stream>>>(conf, tsc, tix);
  nms_kernel<<<dim3(NCLS, BATCH), 256, 0, stream>>>(loc, priors, tsc, tix, out);
}